// Attention_50826642981451
// MI455X (gfx1250) — compile-verified
//
#include <hip/hip_runtime.h>
#include <hip/hip_bf16.h>
#include <cmath>

typedef float v2f __attribute__((ext_vector_type(2)));
typedef float v8f __attribute__((ext_vector_type(8)));

#define OMEGA_F (6.28318530717958647692f / 19.0f)
#define SPLIT 8   // waves per block, each takes every SPLIT-th key tile

// ---------------------------------------------------------------------------
// Kernel 1: per-(b, position) precompute of rope'd q (scaled), k, and v.
//   qc/qs = SCALE * a * [cos(theta-phi), sin(theta-phi)]
//   kc/ks =          a * [cos(theta),     sin(theta)]
//   vv    = v_w * x1
// ---------------------------------------------------------------------------
__global__ void attn_prep_kernel(const float* __restrict__ x,
                                 const float* __restrict__ q_phi,
                                 const float* __restrict__ v_w,
                                 float* __restrict__ qc, float* __restrict__ qs,
                                 float* __restrict__ kc, float* __restrict__ ks,
                                 float* __restrict__ vv,
                                 float scale, int BL, int L) {
    int idx = blockIdx.x * blockDim.x + threadIdx.x;
    if (idx >= BL) return;
    int pos = idx % L;
    float x0 = x[2 * idx + 0];
    float x1 = x[2 * idx + 1];
    float a = x0 * rsqrtf(0.5f * x0 * x0 + 1e-6f);   // rms-norm factor (~ +/- sqrt2)

    float theta = (float)pos * OMEGA_F;
    float st, ct;
    __sincosf(theta, &st, &ct);
    float phi = q_phi[0];
    float sp, cp;
    __sincosf(phi, &sp, &cp);
    float cqm = ct * cp + st * sp;   // cos(theta - phi)
    float sqm = st * cp - ct * sp;   // sin(theta - phi)

    qc[idx] = scale * a * cqm;
    qs[idx] = scale * a * sqm;
    kc[idx] = a * ct;
    ks[idx] = a * st;
    vv[idx] = v_w[0] * x1;
}

// ---------------------------------------------------------------------------
// Kernel 2: flash-style attention with split-K across waves.
// One block (8 waves) owns a 16-query tile; wave w streams key tiles
// j0 = 16*w, 16*w + 128, ... (32 tiles each). S-tile via
// V_WMMA_F32_16X16X4_F32 (K=2 padded to 4 by zeroing the lanes-16..31
// halves of A/B per the ISA 32-bit 16x4 operand layout). Softmax uses a
// fixed upper bound mb = 2*SCALE (logits provably bounded), so partials
// are purely additive -> cross-wave combine is just ds_add_f32 in LDS.
// ---------------------------------------------------------------------------
__global__ __launch_bounds__(256) void
attn_flash_kernel(const float* __restrict__ mask,
                  const float* __restrict__ qc, const float* __restrict__ qs,
                  const float* __restrict__ kc, const float* __restrict__ ks,
                  const float* __restrict__ vv,
                  float* __restrict__ out,
                  float mb, int L) {
    __shared__ float sl[16];   // sum(p)   per query row of this tile
    __shared__ float so[16];   // sum(p*v) per query row of this tile

    const int lane = threadIdx.x & 31;
    const int wave = threadIdx.x >> 5;               // split index 0..7
    const int tile = blockIdx.x;                     // one 16-query tile/block
    const int tilesPerB = L >> 4;
    const int b    = tile / tilesPerB;
    const int i0   = (tile % tilesPerB) << 4;        // first query row of tile
    const int l16  = lane & 15;
    const int half = lane >> 4;
    const int base = b * L;

    if (threadIdx.x < 16) { sl[threadIdx.x] = 0.0f; so[threadIdx.x] = 0.0f; }
    __syncthreads();

    // A operand: 16x4 f32. Lanes 0-15 hold (M=lane, K=0..1); lanes 16-31 hold
    // K=2..3 which are the zero padding of our K=2 problem.
    float aqc = qc[base + i0 + l16];
    float aqs = qs[base + i0 + l16];
    v2f A = { half ? 0.0f : aqc, half ? 0.0f : aqs };

    float lsum[8], osum[8];
#pragma unroll
    for (int r = 0; r < 8; ++r) { lsum[r] = 0.0f; osum[r] = 0.0f; }

    for (int j0 = wave * 16; j0 < L; j0 += 16 * SPLIT) {
        // B operand: 4x16 f32. Lanes 0-15: rows K=0..1 (cols N=lane);
        // lanes 16-31: rows K=2..3 -> zero padding.
        float bkc = kc[base + j0 + l16];
        float bks = ks[base + j0 + l16];
        v2f Bm = { half ? 0.0f : bkc, half ? 0.0f : bks };

        v8f C = {};
        // D[m][n] = SCALE * (q_m . k_n); D layout: VGPR r -> rows r (lanes 0-15)
        // and r+8 (lanes 16-31), column = lane & 15.
        v8f S = __builtin_amdgcn_wmma_f32_16x16x4_f32(
            /*neg_a=*/false, A, /*neg_b=*/false, Bm,
            /*c_mod=*/(short)0, C, /*reuse_a=*/false, /*reuse_b=*/false);

        float vvv = vv[base + j0 + l16];
        const float* mrow = mask + (size_t)(i0 + half * 8) * (size_t)L + (j0 + l16);
#pragma unroll
        for (int r = 0; r < 8; ++r) {
            float s = S[r] + mrow[(size_t)r * (size_t)L];
            float p = __expf(s - mb);                // v_exp_f32, TRANS pipe
            lsum[r] += p;
            osum[r] += p * vvv;
        }
    }

    // 16-lane butterfly reduction per row (stays within each half-wave),
    // then additive combine across the 8 split waves via LDS atomics.
#pragma unroll
    for (int r = 0; r < 8; ++r) {
        float l = lsum[r];
        float o = osum[r];
#pragma unroll
        for (int m = 1; m < 16; m <<= 1) {
            l += __shfl_xor(l, m, 32);
            o += __shfl_xor(o, m, 32);
        }
        if (l16 == 0) {
            int row16 = r + half * 8;                // row within tile
            atomicAdd(&sl[row16], l);                // ds_add_f32
            atomicAdd(&so[row16], o);
        }
    }
    __syncthreads();

    if (threadIdx.x < 16) {
        int row = i0 + threadIdx.x;
        size_t oi = ((size_t)base + row) * 2;
        out[oi + 0] = 0.0f;                          // o_proj channel swap: d0 = 0
        out[oi + 1] = so[threadIdx.x] / sl[threadIdx.x];
    }
}

// ---------------------------------------------------------------------------
extern "C" void kernel_launch(void* const* d_in, const int* in_sizes, int n_in,
                              void* d_out, int out_size, void* d_ws, size_t ws_size,
                              hipStream_t stream) {
    const float* x     = (const float*)d_in[0];   // (B, L, 2)
    const float* mask  = (const float*)d_in[1];   // (1, 1, L, L)
    const float* q_phi = (const float*)d_in[2];   // (1,)
    const float* v_w   = (const float*)d_in[3];   // (1,)

    const int LL = in_sizes[1];
    const int L  = (int)llround(sqrt((double)LL));   // 4096
    const int BL = in_sizes[0] / 2;                  // B*L

    float* ws = (float*)d_ws;
    float* qc = ws + 0 * (size_t)BL;
    float* qs = ws + 1 * (size_t)BL;
    float* kc = ws + 2 * (size_t)BL;
    float* ks = ws + 3 * (size_t)BL;
    float* vv = ws + 4 * (size_t)BL;

    const double omega = 2.0 * M_PI / 19.0;
    const double amp   = log(10.0) / (cos(omega * 0.3) - cos(omega * 0.7));
    const double scale = 0.5 * amp;       // HEAD_DIM^-0.5 * QK_NORM_SCALE^2
    const float  mb    = (float)(2.0 * scale);   // provable logit upper bound

    attn_prep_kernel<<<(BL + 255) / 256, 256, 0, stream>>>(
        x, q_phi, v_w, qc, qs, kc, ks, vv, (float)scale, BL, L);

    const int ntiles = BL / 16;                  // one 16-query tile per block
    attn_flash_kernel<<<ntiles, 256, 0, stream>>>(
        mask, qc, qs, kc, ks, vv, (float*)d_out, mb, L);
}